// OptNetEq_41970420417255
// MI455X (gfx1250) — compile-verified
//
#include <hip/hip_runtime.h>
#include <math.h>

// ---------------- problem constants (match reference) ----------------
#define NX     729          // decision vars
#define MEQ    324          // equality constraints
#define BATCH  64
#define QPEN   0.1f
#define SIGMA  0.1f
#define NITERS 30
#define TRI    ((MEQ*(MEQ+1))/2)   // 52650 packed lower-triangle elements

typedef float v2f __attribute__((ext_vector_type(2)));
typedef float v8f __attribute__((ext_vector_type(8)));

// ---------------- workspace layout (in floats) ----------------
#define OFF_B     ((size_t)0)                       // MEQ
#define OFF_Z     ((size_t)512)                     // BATCH*NX
#define OFF_LAM   (OFF_Z   + (size_t)BATCH*NX)
#define OFF_NU    (OFF_LAM + (size_t)BATCH*NX)      // BATCH*MEQ
#define OFF_D     (OFF_NU  + (size_t)BATCH*MEQ)
#define OFF_G     (OFF_D   + (size_t)BATCH*NX)
#define OFF_RC    (OFF_G   + (size_t)BATCH*NX)
#define OFF_RHS   (OFF_RC  + (size_t)BATCH*NX)      // BATCH*MEQ
#define OFF_DNU   (OFF_RHS + (size_t)BATCH*MEQ)     // BATCH*MEQ
#define OFF_ADAT  (OFF_DNU + (size_t)BATCH*MEQ)     // BATCH*MEQ*MEQ

// =====================================================================
// init: z=1, lam=1, nu=0; block 0 also computes bvec = A @ exp(log_z0)
// =====================================================================
__global__ void oqp_init(const float* __restrict__ A,
                         const float* __restrict__ log_z0,
                         float* __restrict__ bvec,
                         float* __restrict__ z,
                         float* __restrict__ lam,
                         float* __restrict__ nu) {
    const int b = blockIdx.x, t = threadIdx.x;
    for (int i = t; i < NX; i += 256) { z[b*NX+i] = 1.0f; lam[b*NX+i] = 1.0f; }
    for (int i = t; i < MEQ; i += 256) nu[b*MEQ+i] = 0.0f;
    if (b == 0) {
        __shared__ float s_e[NX];
        for (int i = t; i < NX; i += 256) s_e[i] = expf(log_z0[i]);
        __syncthreads();
        for (int mm = t; mm < MEQ; mm += 256) {
            const float* Ar = A + (size_t)mm*NX;
            float s = 0.0f;
            for (int k = 0; k < NX; ++k) s += Ar[k]*s_e[k];
            bvec[mm] = s;
        }
    }
}

// =====================================================================
// resid: per-batch mu, d, g, rc, rhs.  One block (256 thr) per batch.
// =====================================================================
__global__ void oqp_resid(const float* __restrict__ puzzles,
                          const float* __restrict__ A,
                          const float* __restrict__ bvec,
                          const float* __restrict__ z,
                          const float* __restrict__ lam,
                          const float* __restrict__ nu,
                          float* __restrict__ d,
                          float* __restrict__ g,
                          float* __restrict__ rc,
                          float* __restrict__ rhs) {
    const int b = blockIdx.x, t = threadIdx.x;
    __shared__ float s_nu[MEQ];
    __shared__ float s_z[NX];
    __shared__ float s_dg[NX];
    __shared__ float red[256];

    const float* zb = z   + (size_t)b*NX;
    const float* lb = lam + (size_t)b*NX;

    for (int i = t; i < MEQ; i += 256) s_nu[i] = nu[b*MEQ+i];

    float acc = 0.0f;
    for (int i = t; i < NX; i += 256) { float zi = zb[i]; s_z[i] = zi; acc += zi*lb[i]; }
    red[t] = acc; __syncthreads();
    for (int s = 128; s > 0; s >>= 1) { if (t < s) red[t] += red[t+s]; __syncthreads(); }
    const float sm = SIGMA * (red[0] / (float)NX);

    for (int i = t; i < NX; i += 256) {
        float zi = s_z[i], li = lb[i];
        float dot = 0.0f;
        for (int mm = 0; mm < MEQ; ++mm) dot += s_nu[mm]*A[(size_t)mm*NX + i];
        float rzi = QPEN*zi - puzzles[(size_t)b*NX+i] - li + dot;   // p = -puzzles
        float rci = zi*li - sm;
        float di  = 1.0f/(QPEN + li/zi);
        float gi  = -(rzi + rci/zi);
        d [b*NX+i] = di;
        g [b*NX+i] = gi;
        rc[b*NX+i] = rci;
        s_dg[i] = di*gi;
    }
    __syncthreads();

    for (int mm = t; mm < MEQ; mm += 256) {
        const float* Ar = A + (size_t)mm*NX;
        float s1 = 0.0f, s2 = 0.0f;
        for (int k = 0; k < NX; ++k) { float a = Ar[k]; s1 += a*s_z[k]; s2 += a*s_dg[k]; }
        rhs[b*MEQ+mm] = s2 + s1 - bvec[mm];   // A(d*g) + (A z - b)
    }
}

// =====================================================================
// adat: ADAT[b] = (A*diag(d_b)) @ A^T via V_WMMA_F32_16X16X4_F32.
// One wave32 per 16x16 tile; triangular tile pairs (tn <= tm).
// Only the LOWER triangle of ADAT is stored (chol reads only j<=i).
// grid = (231, BATCH), block = 32.
// =====================================================================
#define LDSW 66   // row stride (floats): conflict-free, 8B-aligned rows
__global__ void oqp_adat(const float* __restrict__ A,
                         const float* __restrict__ d,
                         float* __restrict__ ADAT) {
    __shared__ float sA[16*LDSW];
    __shared__ float sB[16*LDSW];
    const int t = threadIdx.x;            // 0..31 (one wave)
    const int batch = blockIdx.y;
    const int tri = blockIdx.x;           // 0..230 triangular index

    int tm = (int)((sqrtf(8.0f*(float)tri + 1.0f) - 1.0f)*0.5f);
    while ((tm+1)*(tm+2)/2 <= tri) ++tm;
    while (tm*(tm+1)/2 > tri) --tm;
    const int tn = tri - tm*(tm+1)/2;

    const int half = t >> 4;              // 0: K pair base, 1: +2
    const int r    = t & 15;
    const float* dv = d + (size_t)batch*NX;

    v8f c = {};
    for (int kc = 0; kc < NX; kc += 64) {
        for (int idx = t; idx < 16*64; idx += 32) {
            int row = idx >> 6, kk = idx & 63, k = kc + kk;
            float av = 0.0f, bv = 0.0f;
            if (k < NX) {
                float dk = dv[k];
                int ga = tm*16 + row, gb = tn*16 + row;
                if (ga < MEQ) av = A[(size_t)ga*NX + k]*dk;
                if (gb < MEQ) bv = A[(size_t)gb*NX + k];
            }
            sA[row*LDSW + kk] = av;
            sB[row*LDSW + kk] = bv;
        }
        __syncthreads();
        #pragma unroll
        for (int kk = 0; kk < 64; kk += 4) {
            const int k0 = kk + half*2;
            v2f a, bb;
            a.x  = sA[r*LDSW + k0]; a.y  = sA[r*LDSW + k0 + 1];
            bb.x = sB[r*LDSW + k0]; bb.y = sB[r*LDSW + k0 + 1];
            // 8 args: (neg_a, A, neg_b, B, c_mod, C, reuse_a, reuse_b)
            c = __builtin_amdgcn_wmma_f32_16x16x4_f32(
                    false, a, false, bb, (short)0, c, false, false);
        }
        __syncthreads();
    }

    float* Mb = ADAT + (size_t)batch*MEQ*MEQ;
    #pragma unroll
    for (int j = 0; j < 8; ++j) {
        int rr = tm*16 + j + half*8;      // C layout: VGPR j -> M=j (+8 for hi half)
        int cc = tn*16 + r;
        if (rr < MEQ && cc < MEQ && rr >= cc)
            Mb[(size_t)rr*MEQ + cc] = c[j];
    }
}

// =====================================================================
// chol: packed lower triangle (210.6 KB) held ENTIRELY in LDS (CDNA5:
// 320 KB/WGP).  Factor + both triangular solves in LDS; only dnu goes
// back to global.  One block (512 thr) per batch; dynamic LDS = TRI*4 B.
// =====================================================================
__global__ void oqp_chol(const float* __restrict__ ADAT,
                         const float* __restrict__ rhs,
                         float* __restrict__ dnu) {
    extern __shared__ float sL[];         // packed: (i,j) j<=i at i*(i+1)/2+j
    __shared__ float s_col[MEQ];
    __shared__ float s_x[MEQ];
    __shared__ float s_inv;
    const int b = blockIdx.x, t = threadIdx.x;
    const int NT = blockDim.x;            // 512
    const float* Mb = ADAT + (size_t)b*MEQ*MEQ;

    // ---- load packed lower triangle from global ----
    for (int idx = t; idx < TRI; idx += NT) {
        int i = (int)((sqrtf(8.0f*(float)idx + 1.0f) - 1.0f)*0.5f);
        while ((i+1)*(i+2)/2 <= idx) ++i;
        while (i*(i+1)/2 > idx) --i;
        int j = idx - i*(i+1)/2;
        sL[idx] = Mb[(size_t)i*MEQ + j];
    }
    __syncthreads();

    // ---- factor (right-looking) ----
    for (int j = 0; j < MEQ; ++j) {
        const int bj = (j*(j+1))/2;
        if (t == 0) {
            float v = sqrtf(fmaxf(sL[bj + j], 1e-30f));
            sL[bj + j] = v;
            s_inv = 1.0f/v;
        }
        __syncthreads();
        const float inv = s_inv;
        for (int i = j+1+t; i < MEQ; i += NT) {
            float v = sL[(i*(i+1))/2 + j]*inv;
            sL[(i*(i+1))/2 + j] = v;
            s_col[i] = v;
        }
        __syncthreads();
        for (int i = j+1+t; i < MEQ; i += NT) {
            const int bi = (i*(i+1))/2;
            const float lij = s_col[i];
            for (int k = j+1; k <= i; ++k) sL[bi + k] -= lij*s_col[k];
        }
        __syncthreads();
    }

    // ---- forward solve L y = rhs ----
    for (int i = t; i < MEQ; i += NT) s_x[i] = rhs[b*MEQ+i];
    __syncthreads();
    for (int j = 0; j < MEQ; ++j) {
        const int bj = (j*(j+1))/2;
        if (t == 0) s_x[j] /= sL[bj + j];
        __syncthreads();
        const float xj = s_x[j];
        for (int i = j+1+t; i < MEQ; i += NT) s_x[i] -= sL[(i*(i+1))/2 + j]*xj;
        __syncthreads();
    }
    // ---- backward solve L^T x = y ----
    for (int j = MEQ-1; j >= 0; --j) {
        const int bj = (j*(j+1))/2;
        if (t == 0) s_x[j] /= sL[bj + j];
        __syncthreads();
        const float xj = s_x[j];
        for (int i = t; i < j; i += NT) s_x[i] -= sL[bj + i]*xj;   // L[j][i], contiguous
        __syncthreads();
    }
    for (int i = t; i < MEQ; i += NT) dnu[b*MEQ+i] = s_x[i];
}

// =====================================================================
// update: dz, dlam, fraction-to-boundary alpha, state updates.
// One block (256 thr) per batch.
// =====================================================================
#define NXPT 3   // ceil(NX/256)
__global__ void oqp_update(const float* __restrict__ A,
                           float* __restrict__ z,
                           float* __restrict__ lam,
                           float* __restrict__ nu,
                           const float* __restrict__ d,
                           const float* __restrict__ g,
                           const float* __restrict__ rc,
                           const float* __restrict__ dnu) {
    const int b = blockIdx.x, t = threadIdx.x;
    __shared__ float s_dnu[MEQ];
    __shared__ float red[256];

    for (int i = t; i < MEQ; i += 256) s_dnu[i] = dnu[b*MEQ+i];
    __syncthreads();

    float dz_l[NXPT], dl_l[NXPT];
    float amin = 3.0e38f;
    #pragma unroll
    for (int cidx = 0; cidx < NXPT; ++cidx) {
        const int i = t + cidx*256;
        dz_l[cidx] = 0.0f; dl_l[cidx] = 0.0f;
        if (i < NX) {
            float dot = 0.0f;
            for (int mm = 0; mm < MEQ; ++mm) dot += s_dnu[mm]*A[(size_t)mm*NX + i];
            const float zi  = z[b*NX+i], li = lam[b*NX+i];
            const float dzi = d[b*NX+i]*(g[b*NX+i] - dot);
            const float dli = -(rc[b*NX+i] + li*dzi)/zi;
            dz_l[cidx] = dzi; dl_l[cidx] = dli;
            if (dzi < 0.0f) amin = fminf(amin, -zi/dzi);
            if (dli < 0.0f) amin = fminf(amin, -li/dli);
        }
    }
    red[t] = amin; __syncthreads();
    for (int s = 128; s > 0; s >>= 1) { if (t < s) red[t] = fminf(red[t], red[t+s]); __syncthreads(); }
    const float alpha = fminf(1.0f, 0.99f*red[0]);

    #pragma unroll
    for (int cidx = 0; cidx < NXPT; ++cidx) {
        const int i = t + cidx*256;
        if (i < NX) {
            z[b*NX+i]   += alpha*dz_l[cidx];
            lam[b*NX+i] += alpha*dl_l[cidx];
        }
    }
    for (int mm = t; mm < MEQ; mm += 256) nu[b*MEQ+mm] += alpha*s_dnu[mm];
}

// =====================================================================
// out: d_out = z (float32, flat [BATCH,9,9,9])
// =====================================================================
__global__ void oqp_out(const float* __restrict__ z, float* __restrict__ out) {
    const int i = blockIdx.x*blockDim.x + threadIdx.x;
    if (i < BATCH*NX) out[i] = z[i];
}

// =====================================================================
extern "C" void kernel_launch(void* const* d_in, const int* in_sizes, int n_in,
                              void* d_out, int out_size, void* d_ws, size_t ws_size,
                              hipStream_t stream) {
    const float* puzzles = (const float*)d_in[0];   // [64,9,9,9] f32
    const float* A       = (const float*)d_in[1];   // [324,729]  f32
    const float* log_z0  = (const float*)d_in[2];   // [729]      f32
    float* out = (float*)d_out;
    float* ws  = (float*)d_ws;

    float* bvec = ws + OFF_B;
    float* z    = ws + OFF_Z;
    float* lam  = ws + OFF_LAM;
    float* nu   = ws + OFF_NU;
    float* d    = ws + OFF_D;
    float* g    = ws + OFF_G;
    float* rc   = ws + OFF_RC;
    float* rhs  = ws + OFF_RHS;
    float* dnu  = ws + OFF_DNU;
    float* ADAT = ws + OFF_ADAT;

    oqp_init<<<dim3(BATCH), dim3(256), 0, stream>>>(A, log_z0, bvec, z, lam, nu);

    const int NTILE = MEQ/16 + ((MEQ % 16) ? 1 : 0);      // 21
    const int NTRI  = NTILE*(NTILE+1)/2;                  // 231
    const size_t cholLds = (size_t)TRI*sizeof(float);     // 210,600 B < 320 KB/WGP

    for (int it = 0; it < NITERS; ++it) {
        oqp_resid<<<dim3(BATCH), dim3(256), 0, stream>>>(puzzles, A, bvec, z, lam, nu,
                                                         d, g, rc, rhs);
        oqp_adat<<<dim3(NTRI, BATCH), dim3(32), 0, stream>>>(A, d, ADAT);
        oqp_chol<<<dim3(BATCH), dim3(512), cholLds, stream>>>(ADAT, rhs, dnu);
        oqp_update<<<dim3(BATCH), dim3(256), 0, stream>>>(A, z, lam, nu, d, g, rc, dnu);
    }

    oqp_out<<<dim3((BATCH*NX + 255)/256), dim3(256), 0, stream>>>(z, out);
}